// StoppingTimeProximityLoss_75857712381993
// MI455X (gfx1250) — compile-verified
//
#include <hip/hip_runtime.h>
#include <hip/hip_bf16.h>

#define N_ 2048
#define T_ 365
#define C_ 64

typedef __attribute__((ext_vector_type(2))) float v2f;
typedef __attribute__((ext_vector_type(2))) int   v2i;
typedef __attribute__((ext_vector_type(8))) float v8f;

// ---------------------------------------------------------------------------
// Kernel 0: zero the 4 accumulator slots in workspace (graph-capture safe).
// ---------------------------------------------------------------------------
__global__ void stpl_init_kernel(float* __restrict__ sums) {
    if (threadIdx.x < 4) sums[threadIdx.x] = 0.0f;
}

// ---------------------------------------------------------------------------
// Kernel 1: one wave32 per row n. Gather logp[n, t, y], reduce over T,
// find t_stop, emit per-row v / tau / y0 and atomic scalar partials.
// ---------------------------------------------------------------------------
__global__ void stpl_row_kernel(const float* __restrict__ logp,
                                const float* __restrict__ tleft,
                                const int*   __restrict__ ytrue,
                                float* __restrict__ sums,     // [0]=cls [1]=earl [2]=wrong
                                float* __restrict__ v_arr,
                                float* __restrict__ tau_arr,
                                int*   __restrict__ y0_arr) {
    const int lane = threadIdx.x & 31;
    const int wave = threadIdx.x >> 5;
    const int n = blockIdx.x * (blockDim.x >> 5) + wave;
    if (n >= N_) return;

    const float invT = 1.0f / (float)T_;
    const size_t rowL = (size_t)n * T_ * C_;
    const size_t rowT = (size_t)n * T_;

    float cls = 0.0f, earl = 0.0f, wrong = 0.0f;
    int tstop = 0x7fffffff;

    for (int t = lane; t < T_; t += 32) {
        const int   y   = ytrue[rowT + t];
        const float lp  = logp[rowL + (size_t)t * C_ + y];
        const float raw = tleft[rowT + t];
        // prefetch next strided gather line (global_prefetch_b8)
        if (t + 32 < T_) {
            __builtin_prefetch(&logp[rowL + (size_t)(t + 32) * C_ + y], 0, 0);
        }
        const float p   = expf(lp);
        const float tf  = (float)t * invT;
        const float om  = 1.0f - tf;        // 1 - t/T
        const float tl  = raw * invT;       // timestamps_left / T
        cls   += -lp;
        earl  += p * om * (1.0f - tl);
        const float om2 = om * om;
        wrong += (1.0f - p) * om2 * tl * tl;
        if (raw <= 0.0f && t < tstop) tstop = t;
    }

    // wave32 butterfly reductions
    #pragma unroll
    for (int off = 16; off > 0; off >>= 1) {
        cls   += __shfl_xor(cls,   off, 32);
        earl  += __shfl_xor(earl,  off, 32);
        wrong += __shfl_xor(wrong, off, 32);
        const int o = __shfl_xor(tstop, off, 32);
        tstop = (o < tstop) ? o : tstop;
    }

    if (lane == 0) {
        if (tstop > T_ - 1) tstop = T_ - 1;   // no stop reached -> T-1
        atomicAdd(&sums[0], cls);
        atomicAdd(&sums[1], earl);
        atomicAdd(&sums[2], wrong);
        const int   y0 = ytrue[rowT];         // y_true[n, 0]
        const float v  = expf(logp[rowL + (size_t)tstop * C_ + y0]);
        v_arr[n]   = v;
        tau_arr[n] = (float)tstop * invT;
        y0_arr[n]  = y0;
    }
}

// ---------------------------------------------------------------------------
// Kernel 2: segment sums S0/S1/S2 over 64 classes as a one-hot GEMM via
// V_WMMA_F32_16X16X4_F32 (64x2048 one-hot @ 2048x3), then proximity +
// final scalar combine.
//   S_k[c] = sum_n 1[y0[n]==c] * v[n] * tau[n]^k
// Branch-free inner loop: B coefficients staged in LDS as bco[4][N] with
// bco[3][*]=0 so lanes m>=3 read zeros; A is a v_cmp/v_cndmask pair.
// 4 waves = 4 class tiles; unroll x4 -> 4 back-to-back WMMAs per iter.
// ---------------------------------------------------------------------------
__global__ void stpl_finalize_kernel(const float* __restrict__ sums,
                                     const float* __restrict__ v_arr,
                                     const float* __restrict__ tau_arr,
                                     const int*   __restrict__ y0_arr,
                                     float* __restrict__ out) {
    __shared__ __align__(16) int   y0s[N_];
    __shared__ __align__(16) float bco[4][N_];
    __shared__ float S[3][C_];
    __shared__ float red[4];

    // Stage per-n data into LDS once (shared by all 4 waves / lane groups).
    for (int i = threadIdx.x; i < N_; i += blockDim.x) {
        const float v = v_arr[i];
        const float t = tau_arr[i];
        y0s[i]    = y0_arr[i];
        bco[0][i] = v;
        bco[1][i] = v * t;
        bco[2][i] = v * t * t;
        bco[3][i] = 0.0f;
    }
    __syncthreads();

    const int lane = threadIdx.x & 31;
    const int wave = threadIdx.x >> 5;            // class tile 0..3
    const int m    = lane & 15;                   // A row / B col within tile
    const int kb   = (lane >> 4) * 2;             // K slots: {0,1} or {2,3}
    const int myClass = wave * 16 + m;
    const int mClamp  = (m < 3) ? m : 3;          // cols >=3 -> zero row
    const float* bptr = &bco[mClamp][0];

    v8f acc = {};
    for (int n0 = 0; n0 < N_; n0 += 16) {
        #pragma unroll
        for (int u = 0; u < 4; ++u) {
            const int base = n0 + 4 * u + kb;     // even -> 8B aligned
            const v2i yp = *(const v2i*)(y0s + base);   // ds_load_b64
            const v2f bp = *(const v2f*)(bptr + base);  // ds_load_b64
            v2f A, B;
            A.x = (yp.x == myClass) ? 1.0f : 0.0f;
            A.y = (yp.y == myClass) ? 1.0f : 0.0f;
            B.x = bp.x;
            B.y = bp.y;
            acc = __builtin_amdgcn_wmma_f32_16x16x4_f32(
                      /*neg_a=*/false, A, /*neg_b=*/false, B,
                      /*c_mod=*/(short)0, acc, /*reuse_a=*/false, /*reuse_b=*/false);
        }
    }

    // D layout (16x16 f32): VGPR r = row r (lanes 0-15) / row r+8 (lanes 16-31),
    // column = lane & 15. Spill columns 0..2 (= S0,S1,S2) to LDS.
    const int col    = lane & 15;
    const int rowOff = (lane >> 4) * 8;
    if (col < 3) {
        #pragma unroll
        for (int r = 0; r < 8; ++r) {
            S[col][wave * 16 + rowOff + r] = acc[r];
        }
    }
    __syncthreads();

    // proximity = sum_c S0*S2 - S1^2  (64 classes, first 64 threads)
    float prox = 0.0f;
    if (threadIdx.x < C_) {
        const int c = threadIdx.x;
        prox = S[0][c] * S[2][c] - S[1][c] * S[1][c];
    }
    #pragma unroll
    for (int off = 16; off > 0; off >>= 1) prox += __shfl_xor(prox, off, 32);
    if (lane == 0) red[wave] = prox;
    __syncthreads();

    if (threadIdx.x == 0) {
        const float prox_total = red[0] + red[1] + red[2] + red[3];
        const float invN  = 1.0f / (float)N_;
        const float cls   = sums[0] * invN;   // .sum(1).mean(0)
        const float earl  = sums[1] * invN;
        const float wrong = sums[2] * invN;
        out[0] = 0.25f * cls - 0.25f * earl - 0.25f * wrong + 0.25f * prox_total;
    }
}

// ---------------------------------------------------------------------------
extern "C" void kernel_launch(void* const* d_in, const int* in_sizes, int n_in,
                              void* d_out, int out_size, void* d_ws, size_t ws_size,
                              hipStream_t stream) {
    const float* logp  = (const float*)d_in[0];  // (N,T,C) f32
    const float* tleft = (const float*)d_in[1];  // (N,T)   f32
    const int*   ytrue = (const int*)  d_in[2];  // (N,T)   i32
    float* out = (float*)d_out;

    // workspace layout: [4 sums][2048 v][2048 tau][2048 y0]
    float* ws    = (float*)d_ws;
    float* sums  = ws;
    float* v_arr = ws + 4;
    float* tau_a = ws + 4 + N_;
    int*   y0_a  = (int*)(ws + 4 + 2 * N_);

    stpl_init_kernel<<<1, 32, 0, stream>>>(sums);

    // 8 waves/block, one row per wave -> 256 blocks
    stpl_row_kernel<<<N_ / 8, 256, 0, stream>>>(logp, tleft, ytrue,
                                                sums, v_arr, tau_a, y0_a);

    stpl_finalize_kernel<<<1, 128, 0, stream>>>(sums, v_arr, tau_a, y0_a, out);
}